// SqueezeEmbedding_18846316495093
// MI455X (gfx1250) — compile-verified
//
#include <hip/hip_runtime.h>

// Problem constants from the reference: x is (B, S, D) float32.
#define B_ 16
#define S_ 4096          // power of two -> cheap bit-twiddled index math
#define D_ 1024
#define S_MASK (S_ - 1)
#define S_SHIFT 12       // log2(S_)
#define ROW_F4 (D_ / 4)  // 256 float4 per (b,t) row
#define ROW_SHIFT 8      // log2(ROW_F4)

typedef float v4f __attribute__((ext_vector_type(4)));
typedef int   v4i __attribute__((vector_size(16)));   // matches builtin proto

#define AS_GLOBAL __attribute__((address_space(1)))
#define AS_LOCAL  __attribute__((address_space(3)))

// ---------------------------------------------------------------------------
// Kernel 1: lengths[b] = sum_t (mask[b,t] != 0).  16 blocks x 256 threads.
// The 16 KiB mask row is staged into LDS with the CDNA5 async copy path
// (GLOBAL_LOAD_ASYNC_TO_LDS_B128, ASYNCcnt) when the builtin is available;
// the reduction then runs out of LDS.  256 KiB total traffic — negligible
// next to the 512 MiB main stream.
// ---------------------------------------------------------------------------
__global__ void squeeze_lengths_kernel(const int* __restrict__ mask,
                                       int* __restrict__ lengths) {
    __shared__ int smask[S_];   // 16 KiB staged mask row
    __shared__ int red[256];
    const int b   = blockIdx.x;
    const int tid = threadIdx.x;

#if defined(__gfx1250__) &&                                                   \
    __has_builtin(__builtin_amdgcn_global_load_async_to_lds_b128) &&          \
    __has_builtin(__builtin_amdgcn_s_wait_asynccnt)
    {
        // 4 async B128 transfers per lane: 256 lanes * 16 B * 4 = 16 KiB row.
        const AS_GLOBAL char* g = (const AS_GLOBAL char*)(mask + b * S_);
        AS_LOCAL char*       l  = (AS_LOCAL char*)smask;
        const int byteoff = tid * 16;
#pragma unroll
        for (int k = 0; k < 4; ++k) {
            __builtin_amdgcn_global_load_async_to_lds_b128(
                (AS_GLOBAL v4i*)(g + byteoff + k * 4096),
                (AS_LOCAL  v4i*)(l + byteoff + k * 4096),
                /*imm offset*/ 0, /*cpol*/ 0);
        }
        // Each wave drains its own ASYNCcnt before arriving at the barrier,
        // so after __syncthreads() the whole row is visible in LDS.
        __builtin_amdgcn_s_wait_asynccnt(0);
    }
#else
#pragma unroll
    for (int t = tid; t < S_; t += 256)
        smask[t] = mask[b * S_ + t];
#endif
    __syncthreads();

    int acc = 0;
#pragma unroll
    for (int t = tid; t < S_; t += 256)
        acc += (smask[t] != 0) ? 1 : 0;

    red[tid] = acc;
    __syncthreads();
#pragma unroll
    for (int off = 128; off > 0; off >>= 1) {
        if (tid < off) red[tid] += red[tid + off];
        __syncthreads();
    }
    if (tid == 0) lengths[b] = red[0];
}

// ---------------------------------------------------------------------------
// Kernel 2: streaming masked scale.  One float4 per lane per trip,
// grid-stride, unrolled so the backend batches B128 loads.  NT hints keep
// the 512 MiB stream from thrashing the 192 MB L2; prefetch runs one stride
// ahead (speculative global_prefetch_b8 — OOB addresses silently dropped).
// ---------------------------------------------------------------------------
__global__ void squeeze_apply_kernel(const v4f* __restrict__ x,
                                     const int* __restrict__ mask,
                                     const int* __restrict__ lengths,
                                     v4f* __restrict__ out) {
    const unsigned total  = (unsigned)B_ * S_ * ROW_F4;            // 16M float4
    const unsigned stride = gridDim.x * blockDim.x;
    unsigned i = blockIdx.x * blockDim.x + threadIdx.x;

#pragma unroll 4
    for (; i < total; i += stride) {
        // gfx1250 speculative prefetch of the next grid-stride trip.
        __builtin_prefetch(&x[i + stride], 0, 1);

        const unsigned row = i >> ROW_SHIFT;      // flat (b,t)
        const unsigned t   = row & S_MASK;
        const unsigned b   = row >> S_SHIFT;

        // Uniform across the wave (32 lanes sit inside one 256-float4 row):
        // both loads broadcast from WGP$/L2.
        const int m   = mask[row];
        const int len = lengths[b];
        const float eff = (m != 0 && (int)t < len) ? 1.0f : 0.0f;

        v4f v = __builtin_nontemporal_load(&x[i]);   // global_load_b128, TH=NT
        v *= eff;
        __builtin_nontemporal_store(v, &out[i]);     // global_store_b128, TH=NT
    }
}

// ---------------------------------------------------------------------------
// Host-side launcher.
// ---------------------------------------------------------------------------
extern "C" void kernel_launch(void* const* d_in, const int* in_sizes, int n_in,
                              void* d_out, int out_size, void* d_ws, size_t ws_size,
                              hipStream_t stream) {
    const float* x    = (const float*)d_in[0];   // (B, S, D) float32
    const int*   mask = (const int*)d_in[1];     // (B, S) bool -> int32 per harness
    float*       out  = (float*)d_out;
    int*         lengths = (int*)d_ws;           // 16 ints of scratch

    (void)in_sizes; (void)n_in; (void)out_size; (void)ws_size;

    // Pass 1: per-batch valid lengths (async-staged mask row in LDS).
    squeeze_lengths_kernel<<<B_, 256, 0, stream>>>(mask, lengths);

    // Pass 2: streaming masked scale.  16M float4 total; 8192 blocks x 256
    // threads -> 8 trips/thread, 4-deep unroll for outstanding B128 loads.
    const int threads = 256;
    const int blocks  = 8192;
    squeeze_apply_kernel<<<blocks, threads, 0, stream>>>(
        (const v4f*)x, mask, lengths, (v4f*)out);
}